// QiskitHead_23055384445011
// MI455X (gfx1250) — compile-verified
//
#include <hip/hip_runtime.h>
#include <math.h>

typedef float v2f __attribute__((ext_vector_type(2)));
typedef float v8f __attribute__((ext_vector_type(8)));

#define BATCH 8192
#define FDIM  512
#define NQ    8
#define NSTATE 256
#define NCLS  10

// ---------------------------------------------------------------------------
// Kernel 1: angles = tanh(x @ proj_w^T) * pi/2     (8192x512)x(512x8)
// One wave per batch row; proj_w staged in LDS; butterfly reduction.
// ---------------------------------------------------------------------------
__global__ void k1_angles(const float* __restrict__ x,
                          const float* __restrict__ proj_w,
                          float* __restrict__ angles) {
    __shared__ float w[NQ * FDIM]; // 16 KB
    int tid = threadIdx.x;
    for (int i = tid; i < NQ * FDIM; i += 256) w[i] = proj_w[i];
    __syncthreads();

    int lane = tid & 31;
    int wave = tid >> 5;
    int row  = blockIdx.x * 8 + wave;
    const float* xr = x + (size_t)row * FDIM;

    float acc[NQ];
#pragma unroll
    for (int q = 0; q < NQ; ++q) acc[q] = 0.0f;

    for (int j = 0; j < FDIM / 32; ++j) {
        float xv = xr[lane + 32 * j];
#pragma unroll
        for (int q = 0; q < NQ; ++q)
            acc[q] += xv * w[q * FDIM + lane + 32 * j];
    }
#pragma unroll
    for (int q = 0; q < NQ; ++q) {
        float v = acc[q];
#pragma unroll
        for (int off = 16; off >= 1; off >>= 1) v += __shfl_xor(v, off, 32);
        if (lane == q)
            angles[row * NQ + q] = tanhf(v) * 1.5707963267948966f;
    }
}

// ---------------------------------------------------------------------------
// Kernel 2: build fixed 256x256 complex unitary U of the parameterized layers
// (post-encoding part of the circuit). One wave per column t: statevector of
// 256 complex amps lives in registers (8 per lane, real+imag). Qubits 0-4 map
// to lane bits (shuffles), qubits 5-7 map to register-index bits.
// Output layout (K-pair interleaved for float2 B-fragment loads):
//   UP[(t>>1)*512 + s*2 + (t&1)] = U[s][t]
// ---------------------------------------------------------------------------
__device__ __forceinline__ void rot_lane_rx(float sr[8], float si[8],
                                            float c, float s, int m) {
#pragma unroll
    for (int r = 0; r < 8; ++r) {
        float pr = __shfl_xor(sr[r], m, 32);
        float pi = __shfl_xor(si[r], m, 32);
        float nr = c * sr[r] + s * pi;
        float ni = c * si[r] - s * pr;
        sr[r] = nr; si[r] = ni;
    }
}

__device__ __forceinline__ void rot_lane_ry(float sr[8], float si[8],
                                            float c, float s, int m, int lane) {
    float b = (lane & m) ? s : -s;
#pragma unroll
    for (int r = 0; r < 8; ++r) {
        float pr = __shfl_xor(sr[r], m, 32);
        float pi = __shfl_xor(si[r], m, 32);
        sr[r] = c * sr[r] + b * pr;
        si[r] = c * si[r] + b * pi;
    }
}

__device__ __forceinline__ void rot_reg_rx(float sr[8], float si[8],
                                           float c, float s, int rm) {
#pragma unroll
    for (int r0 = 0; r0 < 8; ++r0) {
        if (r0 & rm) continue;
        int r1 = r0 | rm;
        float a0r = sr[r0], a0i = si[r0], a1r = sr[r1], a1i = si[r1];
        sr[r0] = c * a0r + s * a1i;  si[r0] = c * a0i - s * a1r;
        sr[r1] = c * a1r + s * a0i;  si[r1] = c * a1i - s * a0r;
    }
}

__device__ __forceinline__ void rot_reg_ry(float sr[8], float si[8],
                                           float c, float s, int rm) {
#pragma unroll
    for (int r0 = 0; r0 < 8; ++r0) {
        if (r0 & rm) continue;
        int r1 = r0 | rm;
        float a0r = sr[r0], a0i = si[r0], a1r = sr[r1], a1i = si[r1];
        sr[r0] = c * a0r - s * a1r;  si[r0] = c * a0i - s * a1i;
        sr[r1] = s * a0r + c * a1r;  si[r1] = s * a0i + c * a1i;
    }
}

__device__ __forceinline__ void cnot_ll(float sr[8], float si[8],
                                        int cm, int tm, int lane) {
    bool hi = (lane & cm) != 0;
#pragma unroll
    for (int r = 0; r < 8; ++r) {
        float pr = __shfl_xor(sr[r], tm, 32);
        float pi = __shfl_xor(si[r], tm, 32);
        sr[r] = hi ? pr : sr[r];
        si[r] = hi ? pi : si[r];
    }
}

__device__ __forceinline__ void cnot_lane_reg(float sr[8], float si[8],
                                              int cm, int trm, int lane) {
    bool hi = (lane & cm) != 0;
#pragma unroll
    for (int r0 = 0; r0 < 8; ++r0) {
        if (r0 & trm) continue;
        int r1 = r0 | trm;
        float t0r = sr[r0], t0i = si[r0];
        sr[r0] = hi ? sr[r1] : sr[r0];  si[r0] = hi ? si[r1] : si[r0];
        sr[r1] = hi ? t0r    : sr[r1];  si[r1] = hi ? t0i    : si[r1];
    }
}

__device__ __forceinline__ void cnot_reg_lane(float sr[8], float si[8],
                                              int crm, int tm) {
#pragma unroll
    for (int r = 0; r < 8; ++r) {
        if (!(r & crm)) continue;
        sr[r] = __shfl_xor(sr[r], tm, 32);
        si[r] = __shfl_xor(si[r], tm, 32);
    }
}

__device__ __forceinline__ void cnot_reg_reg(float sr[8], float si[8],
                                             int crm, int trm) {
#pragma unroll
    for (int r0 = 0; r0 < 8; ++r0) {
        if (!(r0 & crm) || (r0 & trm)) continue;
        int r1 = r0 | trm;
        float tr = sr[r0], ti = si[r0];
        sr[r0] = sr[r1]; si[r0] = si[r1];
        sr[r1] = tr;     si[r1] = ti;
    }
}

__global__ void k2_unitary(const float* __restrict__ qnn_w,
                           float* __restrict__ UrP,
                           float* __restrict__ UiP) {
    int lane = threadIdx.x & 31;
    int t    = blockIdx.x * 8 + (threadIdx.x >> 5); // column index 0..255

    float sr[8], si[8];
#pragma unroll
    for (int r = 0; r < 8; ++r) {
        sr[r] = (r * 32 + lane == t) ? 1.0f : 0.0f;
        si[r] = 0.0f;
    }

#pragma unroll
    for (int l = 0; l < 3; ++l) {
#pragma unroll
        for (int q = 0; q < NQ; ++q) {
            float w0 = qnn_w[(l * NQ + q) * 3 + 0] * 0.5f;
            float w1 = qnn_w[(l * NQ + q) * 3 + 1] * 0.5f;
            float w2 = qnn_w[(l * NQ + q) * 3 + 2] * 0.5f;
            float c0 = cosf(w0), s0 = sinf(w0);
            float c1 = cosf(w1), s1 = sinf(w1);
            float c2 = cosf(w2), s2 = sinf(w2);
            if (q < 5) {
                int m = 1 << q;
                rot_lane_rx(sr, si, c0, s0, m);
                rot_lane_ry(sr, si, c1, s1, m, lane);
                rot_lane_rx(sr, si, c2, s2, m);
            } else {
                int rm = 1 << (q - 5);
                rot_reg_rx(sr, si, c0, s0, rm);
                rot_reg_ry(sr, si, c1, s1, rm);
                rot_reg_rx(sr, si, c2, s2, rm);
            }
        }
        // CNOT chain: (0,1)(1,2)(2,3)(3,4)(4,5)(5,6)(6,7)(7,0)
        cnot_ll(sr, si, 1, 2, lane);
        cnot_ll(sr, si, 2, 4, lane);
        cnot_ll(sr, si, 4, 8, lane);
        cnot_ll(sr, si, 8, 16, lane);
        cnot_lane_reg(sr, si, 16, 1, lane); // c=q4(lane bit4), t=q5(reg bit0)
        cnot_reg_reg(sr, si, 1, 2);         // c=q5, t=q6
        cnot_reg_reg(sr, si, 2, 4);         // c=q6, t=q7
        cnot_reg_lane(sr, si, 4, 1);        // c=q7(reg bit2), t=q0(lane bit0)
    }

#pragma unroll
    for (int r = 0; r < 8; ++r) {
        int s = r * 32 + lane;
        int off = (t >> 1) * (2 * NSTATE) + s * 2 + (t & 1);
        UrP[off] = sr[r];
        UiP[off] = si[r];
    }
}

// ---------------------------------------------------------------------------
// Kernel 3 (dominant, WMMA): per 16-row batch tile:
//   enc[16][256] from angles (separable RY encoding, real)
//   psi = enc @ U^T  via V_WMMA_F32_16X16X4_F32 (real & imag GEMMs)
//   p = |psi|^2 ; E_q = sum_s p*(+/-1) ; out = E @ out_w^T + out_b
// 512 threads = 16 waves; wave w owns N-tile s in [16w, 16w+16).
// ---------------------------------------------------------------------------
__global__ __launch_bounds__(512) void k3_fused(
        const float* __restrict__ angles,
        const float* __restrict__ UrP,
        const float* __restrict__ UiP,
        const float* __restrict__ out_w,
        const float* __restrict__ out_b,
        float* __restrict__ out) {
    __shared__ __align__(16) float cs[16][NQ][2];
    __shared__ __align__(16) float enc[16][NSTATE];   // 16 KB
    __shared__ __align__(16) float pq[16][NSTATE];    // 16 KB
    __shared__ __align__(16) float E[16][NQ];

    int tid  = threadIdx.x;
    int lane = tid & 31;
    int wave = tid >> 5;
    int row0 = blockIdx.x * 16;

    if (tid < 128) {
        int m = tid >> 3, q = tid & 7;
        float a = angles[(row0 + m) * NQ + q] * 0.5f;
        cs[m][q][0] = cosf(a);
        cs[m][q][1] = sinf(a);
    }
    __syncthreads();

#pragma unroll
    for (int e = 0; e < 8; ++e) {
        int idx = tid + 512 * e;
        int m = idx >> 8, t = idx & 255;
        float v = 1.0f;
#pragma unroll
        for (int q = 0; q < NQ; ++q) v *= cs[m][q][(t >> q) & 1];
        enc[m][t] = v;
    }
    __syncthreads();

    // --- WMMA GEMM: 64 K-steps of 16x16x4 fp32, real + imag ---
    v8f cr, ci;
#pragma unroll
    for (int v = 0; v < 8; ++v) { cr[v] = 0.0f; ci[v] = 0.0f; }

    int mrow = lane & 15;        // A fragment: M = lane%16
    int h    = lane >> 4;        // K-half select
    int n    = lane & 15;        // B/C fragment: N = lane%16
    int s0   = wave * 16;

    for (int k = 0; k < NSTATE; k += 4) {
        int kk = k + 2 * h;                       // even
        v2f a  = *(const v2f*)&enc[mrow][kk];     // enc[M][kk], enc[M][kk+1]
        const float* pr = &UrP[(kk >> 1) * (2 * NSTATE) + (s0 + n) * 2];
        const float* pi = &UiP[(kk >> 1) * (2 * NSTATE) + (s0 + n) * 2];
        v2f br = *(const v2f*)pr;                 // U^T[kk][s], U^T[kk+1][s]
        v2f bi = *(const v2f*)pi;
        cr = __builtin_amdgcn_wmma_f32_16x16x4_f32(false, a, false, br,
                                                   (short)0, cr, false, false);
        ci = __builtin_amdgcn_wmma_f32_16x16x4_f32(false, a, false, bi,
                                                   (short)0, ci, false, false);
    }

    // probabilities |psi|^2 into LDS; C layout: lane l, vgpr v -> M=v+8*(l>>4), N=l&15
#pragma unroll
    for (int v = 0; v < 8; ++v) {
        int m = v + 8 * h;
        pq[m][s0 + n] = cr[v] * cr[v] + ci[v] * ci[v];
    }
    __syncthreads();

    // Z expectations: E[m][q] = sum_s p[m][s] * (bit_q(s) ? -1 : +1)
    if (tid < 128) {
        int m = tid >> 3, q = tid & 7;
        float acc = 0.0f;
        for (int s = 0; s < NSTATE; ++s) {
            float p = pq[m][s];
            acc += ((s >> q) & 1) ? -p : p;
        }
        E[m][q] = acc;
    }
    __syncthreads();

    // head: out[b][c] = sum_q E[b][q]*out_w[c][q] + out_b[c]
    if (tid < 16 * NCLS) {
        int m = tid / NCLS, c = tid % NCLS;
        float acc = out_b[c];
#pragma unroll
        for (int q = 0; q < NQ; ++q) acc += E[m][q] * out_w[c * NQ + q];
        out[(row0 + m) * NCLS + c] = acc;
    }
}

// ---------------------------------------------------------------------------
extern "C" void kernel_launch(void* const* d_in, const int* in_sizes, int n_in,
                              void* d_out, int out_size, void* d_ws, size_t ws_size,
                              hipStream_t stream) {
    const float* x      = (const float*)d_in[0];
    const float* proj_w = (const float*)d_in[1];
    const float* qnn_w  = (const float*)d_in[2];
    const float* out_w  = (const float*)d_in[3];
    const float* out_b  = (const float*)d_in[4];
    float* ws = (float*)d_ws;

    float* angles = ws;                       // 8192*8      = 65536 f
    float* UrP    = ws + 65536;               // 256*256     = 65536 f
    float* UiP    = ws + 65536 + 65536;       // 256*256     = 65536 f

    k1_angles<<<BATCH / 8, 256, 0, stream>>>(x, proj_w, angles);
    k2_unitary<<<NSTATE / 8, 256, 0, stream>>>(qnn_w, UrP, UiP);
    k3_fused<<<BATCH / 16, 512, 0, stream>>>(angles, UrP, UiP, out_w, out_b,
                                             (float*)d_out);
}